// ExpertChoiceGating_57475252355399
// MI455X (gfx1250) — compile-verified
//
#include <hip/hip_runtime.h>
#include <hip/hip_bf16.h>
#include <math.h>

// Problem constants (match reference)
#define BB 4
#define NN 2048
#define DD 1024
#define EE 16
#define CAP 256   // max(2048*2/16, 4)

typedef float v2f __attribute__((ext_vector_type(2)));
typedef float v8f __attribute__((ext_vector_type(8)));

// ---------------------------------------------------------------------------
// Kernel 1: zero-fill the 256 MB output (dispatch_st + combine + 2 scalars).
// Vectorized float4 stores; grid-stride.
// ---------------------------------------------------------------------------
__global__ __launch_bounds__(256) void zero_fill_kernel(float* __restrict__ out) {
    const size_t total = 2ull * BB * NN * EE * CAP + 2ull;
    const size_t n4    = total >> 2;          // float4 count
    const size_t tid   = (size_t)blockIdx.x * blockDim.x + threadIdx.x;
    const size_t step  = (size_t)gridDim.x * blockDim.x;
    const float4 z = make_float4(0.f, 0.f, 0.f, 0.f);
    for (size_t i = tid; i < n4; i += step)
        ((float4*)out)[i] = z;
    const size_t tail = n4 << 2;
    if (tid < total - tail)
        out[tail + tid] = 0.0f;
}

// ---------------------------------------------------------------------------
// Kernel 2: affinity logits via V_WMMA_F32_16X16X4_F32.
//   aff[b][e][n] = dot(key[b,n,:], query[e,:]) * exp(-softmax_temp)
// One wave computes a 16-token x 16-expert tile, looping K over D in steps
// of 4.  A-matrix layout (32-bit, 16x4): lanes 0-15 carry K=0,1 (VGPR0,1),
// lanes 16-31 carry K=2,3.  B-matrix (4x16): lane = column (expert), same
// K split.  C/D (16x16 f32, 8 VGPRs): lanes 0-15 -> M=r, lanes 16-31 -> M=r+8.
// Four accumulators break the WMMA->WMMA RAW chain on SRC2/VDST.
// ---------------------------------------------------------------------------
__global__ __launch_bounds__(256) void gate_gemm_wmma_kernel(
    const float* __restrict__ key,     // [B][N][D]
    const float* __restrict__ query,   // [E][D]
    const float* __restrict__ temp,    // [1] log-temperature
    float* __restrict__ aff)           // [B][E][N] (logits out)
{
    const int wave = threadIdx.x >> 5;
    const int lane = threadIdx.x & 31;
    const int tile = blockIdx.x * 8 + wave;      // B*(N/16) = 512 tiles
    const int b    = tile >> 7;                  // tile / (N/16)
    const int m0   = (tile & 127) << 4;          // token tile base
    const int half = lane >> 4;                  // 0: K={0,1}  1: K={2,3}
    const int l15  = lane & 15;

    const float inv_temp = __expf(-temp[0]);     // 1 / exp(softmax_temp)

    const float* arow = key   + ((size_t)(b * NN + m0 + l15)) * DD + 2 * half;
    const float* brow = query + (size_t)l15 * DD + 2 * half;

    v8f acc0 = {}, acc1 = {}, acc2 = {}, acc3 = {};
    #pragma unroll 4
    for (int k = 0; k < DD; k += 16) {
        v2f a0 = *(const v2f*)(arow + k);
        v2f b0 = *(const v2f*)(brow + k);
        v2f a1 = *(const v2f*)(arow + k + 4);
        v2f b1 = *(const v2f*)(brow + k + 4);
        v2f a2 = *(const v2f*)(arow + k + 8);
        v2f b2 = *(const v2f*)(brow + k + 8);
        v2f a3 = *(const v2f*)(arow + k + 12);
        v2f b3 = *(const v2f*)(brow + k + 12);
        acc0 = __builtin_amdgcn_wmma_f32_16x16x4_f32(false, a0, false, b0,
                                                     (short)0, acc0, false, false);
        acc1 = __builtin_amdgcn_wmma_f32_16x16x4_f32(false, a1, false, b1,
                                                     (short)0, acc1, false, false);
        acc2 = __builtin_amdgcn_wmma_f32_16x16x4_f32(false, a2, false, b2,
                                                     (short)0, acc2, false, false);
        acc3 = __builtin_amdgcn_wmma_f32_16x16x4_f32(false, a3, false, b3,
                                                     (short)0, acc3, false, false);
    }
    v8f acc = (acc0 + acc1) + (acc2 + acc3);
    acc = acc * inv_temp;

    // Store: expert e = l15 (row of aff), tokens m0 + half*8 + r, r=0..7
    // contiguous -> one 32-byte-aligned v8f store.
    float* outp = aff + ((size_t)(b * EE + l15)) * NN + m0 + half * 8;
    *(v8f*)outp = acc;
}

// ---------------------------------------------------------------------------
// Kernel 3: softmax over the N=2048 tokens of each (b,e) row, in place.
// One 256-thread block (8 waves) per row; wave shuffle + LDS reduction.
// ---------------------------------------------------------------------------
__global__ __launch_bounds__(256) void softmax_tokens_kernel(float* __restrict__ aff) {
    __shared__ float red[8];
    float* row = aff + (size_t)blockIdx.x * NN;
    const int tid  = threadIdx.x;
    const int wave = tid >> 5;
    const int lane = tid & 31;

    // --- max reduction ---
    float m = -INFINITY;
    for (int n = tid; n < NN; n += 256) m = fmaxf(m, row[n]);
    #pragma unroll
    for (int o = 16; o > 0; o >>= 1) m = fmaxf(m, __shfl_xor(m, o, 32));
    if (lane == 0) red[wave] = m;
    __syncthreads();
    m = red[lane & 7];
    #pragma unroll
    for (int o = 4; o > 0; o >>= 1) m = fmaxf(m, __shfl_xor(m, o, 32));
    m = __shfl(m, 0, 32);
    __syncthreads();

    // --- sum of exp ---
    float s = 0.0f;
    for (int n = tid; n < NN; n += 256) s += __expf(row[n] - m);
    #pragma unroll
    for (int o = 16; o > 0; o >>= 1) s += __shfl_xor(s, o, 32);
    if (lane == 0) red[wave] = s;
    __syncthreads();
    s = red[lane & 7];
    #pragma unroll
    for (int o = 4; o > 0; o >>= 1) s += __shfl_xor(s, o, 32);
    s = __shfl(s, 0, 32);
    const float inv_s = 1.0f / s;

    // --- normalize in place ---
    for (int n = tid; n < NN; n += 256)
        row[n] = __expf(row[n] - m) * inv_s;
}

// ---------------------------------------------------------------------------
// Kernel 4: exact top-k via ranking + scatter.
// rank(n) = #{ j : aff[j] > aff[n]  ||  (aff[j]==aff[n] && j<n) }
// reproduces jax.lax.top_k's descending sort with index tie-break; rank is
// the capacity slot c.  Affinity row staged in 8 KB LDS; each block handles
// 256 tokens of one (b,e) row.
// ---------------------------------------------------------------------------
__global__ __launch_bounds__(256) void topk_scatter_kernel(
    const float* __restrict__ aff,     // [B][E][N] softmaxed
    float* __restrict__ out)           // dispatch_st @0, combine @ B*N*E*CAP
{
    __shared__ float sa[NN];           // 8 KB
    const int pair  = blockIdx.x >> 3;         // b*E + e
    const int chunk = blockIdx.x & 7;
    const int b     = pair >> 4;
    const int e     = pair & 15;

    const float* row = aff + (size_t)pair * NN;
    for (int n = threadIdx.x; n < NN; n += 256) sa[n] = row[n];
    __syncthreads();

    const int   n = chunk * 256 + threadIdx.x;
    const float a = sa[n];
    int rank = 0;
    #pragma unroll 8
    for (int j = 0; j < NN; ++j) {
        float v = sa[j];
        rank += (v > a) | ((v == a) & (j < n));
    }
    if (rank < CAP) {
        const size_t idx = (((size_t)(b * NN + n) * EE + e) * CAP) + (size_t)rank;
        out[idx] = 1.0f;                                       // dispatch_st
        out[(size_t)BB * NN * EE * CAP + idx] = a;             // combine
    }
}

// ---------------------------------------------------------------------------
extern "C" void kernel_launch(void* const* d_in, const int* in_sizes, int n_in,
                              void* d_out, int out_size, void* d_ws, size_t ws_size,
                              hipStream_t stream) {
    const float* key   = (const float*)d_in[0];   // [B][N][D]
    const float* query = (const float*)d_in[1];   // [E][D]
    const float* temp  = (const float*)d_in[2];   // [1]
    float*       out   = (float*)d_out;
    float*       aff   = (float*)d_ws;            // [B][E][N] = 512 KB scratch

    // 1) zero the 256 MB output (store-bandwidth bound; ~11 us at 23.3 TB/s)
    zero_fill_kernel<<<2048, 256, 0, stream>>>(out);

    // 2) WMMA GEMM -> scaled affinity logits [B][E][N]
    gate_gemm_wmma_kernel<<<BB * (NN / 16) / 8, 256, 0, stream>>>(key, query, temp, aff);

    // 3) softmax over tokens, one block per (b,e)
    softmax_tokens_kernel<<<BB * EE, 256, 0, stream>>>(aff);

    // 4) rank-based exact top-k + scatter of the 16384 nonzeros
    topk_scatter_kernel<<<BB * EE * (NN / 256), 256, 0, stream>>>(aff, out);
}